// GaussianNLLLoss2d_72395968741496
// MI455X (gfx1250) — compile-verified
//
#include <hip/hip_runtime.h>
#include <hip/hip_bf16.h>

typedef float v2f __attribute__((ext_vector_type(2)));
typedef float v8f __attribute__((ext_vector_type(8)));

// Pre-load the constant part of the mean NLL:  0.5 * D * log(2*pi) = log(2*pi)
__global__ void nll_init_kernel(float* __restrict__ out) {
    if (threadIdx.x == 0) out[0] = 1.8378770664093453f; // log(2*pi)
}

__device__ __forceinline__ float nll_sample(const float2* __restrict__ mu2,
                                            const float2* __restrict__ x2,
                                            const float4* __restrict__ L4,
                                            int i) {
  float2 m  = mu2[i];            // global_load_b64
  float2 xv = x2[i];             // global_load_b64
  float4 Lv = L4[i];             // global_load_b128: L00, L01, L10, L11
  float d0 = xv.x - m.x;
  float d1 = xv.y - m.y;
  // Lower-triangular 2x2 solve:
  float y0 = d0 / Lv.x;
  float y1 = (d1 - Lv.z * y0) / Lv.w;
  // q + logdet  (constant term folded into init kernel)
  return fmaf(y0, y0, y1 * y1) + 2.0f * __logf(Lv.x * Lv.w);
}

__global__ __launch_bounds__(256) void nll_main_kernel(
    const float* __restrict__ mu, const float* __restrict__ L,
    const float* __restrict__ x, float* __restrict__ out,
    int B, float scale) {
  const int gid    = blockIdx.x * blockDim.x + threadIdx.x;
  const int stride = gridDim.x * blockDim.x;

  const float2* __restrict__ mu2 = reinterpret_cast<const float2*>(mu);
  const float2* __restrict__ x2  = reinterpret_cast<const float2*>(x);
  const float4* __restrict__ L4  = reinterpret_cast<const float4*>(L);

  // Explicit 2x unroll: two independent accumulator chains, no per-iteration
  // selects; acc0/acc1 map directly onto the two A-matrix VGPRs of the WMMA.
  float acc0 = 0.0f, acc1 = 0.0f;
  int i = gid;
  for (; i + stride < B; i += 2 * stride) {
    acc0 += nll_sample(mu2, x2, L4, i);
    acc1 += nll_sample(mu2, x2, L4, i + stride);
  }
  if (i < B) acc0 += nll_sample(mu2, x2, L4, i);

  // WMMA-based wave reduction: D = A(16x4) * ones(4x16) + 0.
  // Sum of ALL 256 entries of D == 16 * sum(all 64 entries of A),
  // independent of the exact lane/VGPR striping. A = the 64 per-lane
  // partial sums of this wave (2 VGPRs x 32 lanes).
  v2f a; a[0] = acc0; a[1] = acc1;
  v2f b; b[0] = 1.0f; b[1] = 1.0f;
  v8f c = {};
  // (neg_a, A, neg_b, B, c_mod, C, reuse_a, reuse_b)
  v8f d = __builtin_amdgcn_wmma_f32_16x16x4_f32(
      false, a, false, b, (short)0, c, false, false);
  float s = ((d[0] + d[1]) + (d[2] + d[3])) + ((d[4] + d[5]) + (d[6] + d[7]));

  // Cross-lane wave32 butterfly: every lane ends with 16 * sum(wave accs)
  for (int off = 16; off > 0; off >>= 1)
    s += __shfl_xor(s, off, 32);

  __shared__ float wsum[8];
  const int lane = threadIdx.x & 31;
  const int wid  = threadIdx.x >> 5;
  if (lane == 0) wsum[wid] = s;
  __syncthreads();
  if (threadIdx.x == 0) {
    float t = 0.0f;
    const int nw = blockDim.x >> 5;
    for (int w = 0; w < nw; ++w) t += wsum[w];
    atomicAdd(out, t * scale);   // scale has the /16 WMMA factor folded in
  }
}

extern "C" void kernel_launch(void* const* d_in, const int* in_sizes, int n_in,
                              void* d_out, int out_size, void* d_ws, size_t ws_size,
                              hipStream_t stream) {
  const float* mu = (const float*)d_in[0];   // [B,2]
  const float* L  = (const float*)d_in[1];   // [B,2,2]
  const float* x  = (const float*)d_in[2];   // [B,2]
  float* out = (float*)d_out;                // scalar mean NLL

  const int B = in_sizes[0] / 2;
  // mean = log(2pi) + (0.5/B) * sum(q + logdet); WMMA path multiplies by 16.
  const float scale = 0.5f / (16.0f * (float)B);

  nll_init_kernel<<<1, 64, 0, stream>>>(out);

  const int threads = 256;                   // 8 wave32 per block
  long long blocks_ll = ((long long)B + threads * 8 - 1) / (threads * 8);
  int blocks = (int)(blocks_ll < 1 ? 1 : (blocks_ll > 4096 ? 4096 : blocks_ll));
  nll_main_kernel<<<blocks, threads, 0, stream>>>(mu, L, x, out, B, scale);
}